// WideComponent_83545703842148
// MI455X (gfx1250) — compile-verified
//
#include <hip/hip_runtime.h>

// Reference: wide-and-cross component.
//   out[i] = emb_user[user_id[i]] + emb_item[item_id[i]] + emb_cat[category[i]]
//          + cross_w[user_id[i]*D_CAT + category[i]]
//          + cross_w[OFF_IC + item_id[i]*D_CAT + category[i]]
// Pure random-gather, latency bound. One row per wave32 lane, all five
// gathers issued before consumption. cross_w (600 MB > 192 MB L2, ~unique
// random indices) uses the gfx1250 non-temporal hint; hot embedding tables
// keep the default temporal policy. Flat cross indices fit in i32
// (max 1.5e8, x4 bytes < 2^31) so all gathers use the scalar-base +
// scaled-VGPR-offset addressing mode (no 64-bit VALU address chains).

namespace {
constexpr int kDCat  = 1000;
constexpr int kOffIC = 100000 * 1000;  // D_USER * D_CAT = 1e8, fits i32
}

__global__ __launch_bounds__(256, 1) void wide_cross_gather(
    const int*   __restrict__ user_id,
    const int*   __restrict__ item_id,
    const int*   __restrict__ category,
    const float* __restrict__ emb_user,
    const float* __restrict__ emb_item,
    const float* __restrict__ emb_cat,
    const float* __restrict__ cross_w,
    float*       __restrict__ out,
    int n)
{
    const int i = blockIdx.x * blockDim.x + threadIdx.x;
    if (i >= n) return;

    // Sequential, coalesced index loads (one s_clause of b32 loads).
    const int u  = user_id[i];
    const int it = item_id[i];
    const int c  = category[i];

    // Hot-table gathers: regular temporal policy (emb_cat is 4 KB, reused
    // ~16x per entry; emb_user/emb_item are L2-resident).
    const float wu = emb_user[u];
    const float wi = emb_item[it];
    const float wc = emb_cat[c];

    // Cold-table gathers: non-temporal so streaming misses don't rinse the
    // hot tables. 32-bit flat indices -> saddr + scale_offset addressing.
    const int iuc = u * kDCat + c;
    const int iic = kOffIC + it * kDCat + c;
    const float xuc = __builtin_nontemporal_load(cross_w + iuc);
    const float xic = __builtin_nontemporal_load(cross_w + iic);

    out[i] = (wu + wi + wc) + (xuc + xic);
}

extern "C" void kernel_launch(void* const* d_in, const int* in_sizes, int n_in,
                              void* d_out, int out_size, void* d_ws, size_t ws_size,
                              hipStream_t stream)
{
    (void)n_in; (void)d_ws; (void)ws_size; (void)out_size;

    const int*   user_id  = (const int*)  d_in[0];
    const int*   item_id  = (const int*)  d_in[1];
    const int*   category = (const int*)  d_in[2];
    const float* emb_user = (const float*)d_in[3];
    const float* emb_item = (const float*)d_in[4];
    const float* emb_cat  = (const float*)d_in[5];
    const float* cross_w  = (const float*)d_in[6];
    float*       out      = (float*)      d_out;

    const int n = in_sizes[0];             // B = 16384 rows
    const int block = 256;                 // 8 wave32s per block
    const int grid  = (n + block - 1) / block;
    wide_cross_gather<<<grid, block, 0, stream>>>(
        user_id, item_id, category, emb_user, emb_item, emb_cat, cross_w, out, n);
}